// VRWKV_SpatialMix_25598005084645
// MI455X (gfx1250) — compile-verified
//
#include <hip/hip_runtime.h>
#include <math.h>

// Problem constants (fixed by the reference: B=8, T=4096, C=192, H=W=64)
#define Bb 8
#define Tt 4096
#define Cc 192
#define HH 64
#define WW 64
#define CHUNK 64
#define NCH (Tt / CHUNK) // 64 chunks per scan chain

typedef float v2f __attribute__((ext_vector_type(2)));
typedef float v8f __attribute__((ext_vector_type(8)));

// Map sequence position -> memory index. perm=1 selects the w-major
// (column) traversal used by the second WKV pass:  s = x*H + y  ->  t = y*W + x.
__device__ __forceinline__ int seq2mem(int s, int perm) {
  return perm ? ((s & (HH - 1)) * WW + (s >> 6)) : s;
}

// ---------------------------------------------------------------------------
// 1) Fold the three depthwise convs + identity into one 5x5 kernel per channel.
//    weff layout: [tap(25)][C] for coalesced reads in the conv kernel.
// ---------------------------------------------------------------------------
__global__ void build_weff(const float* __restrict__ w1, const float* __restrict__ w3,
                           const float* __restrict__ w5, const float* __restrict__ alpha,
                           float* __restrict__ weff) {
  int i = blockIdx.x * blockDim.x + threadIdx.x;
  if (i >= 25 * Cc) return;
  int tap = i / Cc, c = i % Cc;
  int dy = tap / 5, dx = tap % 5;
  float a0 = alpha[0], a1 = alpha[1], a2 = alpha[2], a3 = alpha[3];
  float wv = a3 * w5[c * 25 + tap];
  if (dy >= 1 && dy <= 3 && dx >= 1 && dx <= 3)
    wv += a2 * w3[c * 9 + (dy - 1) * 3 + (dx - 1)];
  if (dy == 2 && dx == 2)
    wv += a1 * w1[c] + a0;
  weff[tap * Cc + c] = wv;
}

// One block per (b, pixel), thread = channel: fully coalesced 768B bursts.
__global__ void omnishift(const float* __restrict__ x, const float* __restrict__ weff,
                          float* __restrict__ xs) {
  const int c = threadIdx.x;
  const int bt = blockIdx.x;
  const int b = bt / Tt, t = bt % Tt;
  const int y = t / WW, xx = t % WW;
  float wreg[25];
#pragma unroll
  for (int i = 0; i < 25; ++i) wreg[i] = weff[i * Cc + c];
  const float* xb = x + (size_t)b * Tt * Cc;
  float acc = 0.f;
#pragma unroll
  for (int dy = 0; dy < 5; ++dy) {
    int iy = y + dy - 2;
    if (iy < 0 || iy >= HH) continue;
#pragma unroll
    for (int dx = 0; dx < 5; ++dx) {
      int ix = xx + dx - 2;
      if (ix < 0 || ix >= WW) continue;
      acc += wreg[dy * 5 + dx] * xb[(size_t)(iy * WW + ix) * Cc + c];
    }
  }
  xs[(size_t)bt * Cc + c] = acc;
}

// ---------------------------------------------------------------------------
// 2) Fused k/v/r projection GEMM on the fp32 matrix pipe.
//    One wave per 16x16 output tile, K swept 4 at a time via
//    V_WMMA_F32_16X16X4_F32; the A fragment is shared by all three products.
//    A-frag (16x4 f32): lanes 0-15 hold K={0,1}, lanes 16-31 K={2,3}.
//    B-frag (4x16 f32): v0 = row 2*half, v1 = row 2*half+1 (B[k][n] = W[n][k]).
// ---------------------------------------------------------------------------
__global__ void gemm_qkv(const float* __restrict__ xs, const float* __restrict__ Wk,
                         const float* __restrict__ Wv, const float* __restrict__ Wr,
                         float* __restrict__ kO, float* __restrict__ vO,
                         float* __restrict__ srO) {
  const int lane = threadIdx.x & 31;
  const int l16 = lane & 15;
  const int half = lane >> 4;
  const int m0 = blockIdx.x * 16;
  const int n0 = blockIdx.y * 16;
  v8f ck = {}; v8f cv = {}; v8f cr = {};
  const size_t aoff = (size_t)(m0 + l16) * Cc + 2 * half;
  const size_t boff = (size_t)(n0 + l16) * Cc + 2 * half;
  for (int kk = 0; kk < Cc; kk += 4) {
    v2f a  = *(const v2f*)(xs + aoff + kk);
    v2f bk = *(const v2f*)(Wk + boff + kk);
    v2f bv = *(const v2f*)(Wv + boff + kk);
    v2f br = *(const v2f*)(Wr + boff + kk);
    ck = __builtin_amdgcn_wmma_f32_16x16x4_f32(false, a, false, bk, (short)0, ck, false, false);
    cv = __builtin_amdgcn_wmma_f32_16x16x4_f32(false, a, false, bv, (short)0, cv, false, false);
    cr = __builtin_amdgcn_wmma_f32_16x16x4_f32(false, a, false, br, (short)0, cr, false, false);
  }
#pragma unroll
  for (int r = 0; r < 8; ++r) {
    size_t idx = (size_t)(m0 + r + 8 * half) * Cc + (n0 + l16);
    kO[idx] = ck[r];
    vO[idx] = cv[r];
    srO[idx] = 1.f / (1.f + __expf(-cr[r]));  // fused sigmoid epilogue
  }
}

// Output GEMM: A = sr .* v fused into the fragment load; D -> d_out.
__global__ void gemm_out(const float* __restrict__ sr, const float* __restrict__ vin,
                         const float* __restrict__ Wo, float* __restrict__ out) {
  const int lane = threadIdx.x & 31;
  const int l16 = lane & 15;
  const int half = lane >> 4;
  const int m0 = blockIdx.x * 16;
  const int n0 = blockIdx.y * 16;
  v8f acc = {};
  const size_t aoff = (size_t)(m0 + l16) * Cc + 2 * half;
  const size_t boff = (size_t)(n0 + l16) * Cc + 2 * half;
  for (int kk = 0; kk < Cc; kk += 4) {
    v2f s2 = *(const v2f*)(sr + aoff + kk);
    v2f w2 = *(const v2f*)(vin + aoff + kk);
    v2f a;
    a.x = s2.x * w2.x;
    a.y = s2.y * w2.y;
    v2f bo = *(const v2f*)(Wo + boff + kk);
    acc = __builtin_amdgcn_wmma_f32_16x16x4_f32(false, a, false, bo, (short)0, acc, false, false);
  }
#pragma unroll
  for (int r = 0; r < 8; ++r)
    out[(size_t)(m0 + r + 8 * half) * Cc + (n0 + l16)] = acc[r];
}

// ---------------------------------------------------------------------------
// 3) WKV: chunked three-pass associative scan in log-scaled (p,q,o) space.
// ---------------------------------------------------------------------------

// Pass 1: per-chunk totals, both directions in parallel.
// sum layout: [dir][b][chunk][{p,q,o}][c]
__global__ void wkv_pass1(const float* __restrict__ k, const float* __restrict__ v,
                          const float* __restrict__ decay, float* __restrict__ sum,
                          int perm) {
  const int c = threadIdx.x;
  const int j = blockIdx.x;
  const int b = blockIdx.y;
  const int dir = blockIdx.z;
  const float w = decay[c] * (1.0f / Tt);
  const size_t bbase = (size_t)b * Tt * Cc;
  float p = 0.f, q = 0.f, o = -1e38f;
  for (int s = 0; s < CHUNK; ++s) {
    int sq = j * CHUNK + s;
    int spos = dir ? (Tt - 1 - sq) : sq;
    int t = seq2mem(spos, perm);
    size_t idx = bbase + (size_t)t * Cc + c;
    float kt = k[idx], vt = v[idx];
    float no = fmaxf(w + o, kt);
    float A = __expf(w + o - no);
    float Bc = __expf(kt - no);
    p = A * p + Bc * vt;
    q = A * q + Bc;
    o = no;
  }
  size_t base = ((((size_t)dir * Bb + b) * NCH + j) * 3) * Cc + c;
  sum[base] = p;
  sum[base + Cc] = q;
  sum[base + 2 * Cc] = o;
}

// Pass 2: exclusive scan over chunk summaries (in place). Combine across a
// chunk of length L applies e^{L*w} decay to the carry.
__global__ void wkv_pass2(const float* __restrict__ decay, float* __restrict__ sum) {
  const int c = threadIdx.x;
  const int b = blockIdx.x;
  const int dir = blockIdx.y;
  const float w = decay[c] * (1.0f / Tt);
  const float Lw = (float)CHUNK * w;
  float p = 0.f, q = 0.f, o = -1e38f;
  const size_t base0 = (((size_t)dir * Bb + b) * NCH) * 3 * Cc + c;
  for (int j = 0; j < NCH; ++j) {
    size_t base = base0 + (size_t)j * 3 * Cc;
    float P = sum[base], Q = sum[base + Cc], O = sum[base + 2 * Cc];
    sum[base] = p;                 // write exclusive carry over the summary
    sum[base + Cc] = q;
    sum[base + 2 * Cc] = o;
    float no = fmaxf(o + Lw, O);
    float A = __expf(o + Lw - no);
    float Bo2 = __expf(O - no);
    p = A * p + Bo2 * P;
    q = A * q + Bo2 * Q;
    o = no;
  }
}

// Pass 3: replay forward states into (thread-private) LDS, then walk the
// mirrored backward chunk, fuse the 3-way log-sum-exp combine, and write y
// in place over v. No __syncthreads needed: LDS columns are per-thread.
__global__ void __launch_bounds__(64) wkv_pass3(const float* __restrict__ k,
                                                float* __restrict__ v,
                                                const float* __restrict__ decay,
                                                const float* __restrict__ first,
                                                const float* __restrict__ sum,
                                                int perm) {
  __shared__ float lds[CHUNK * 3 * 64]; // 48 KB: [s][{p,q,o}][tid]
  const int tid = threadIdx.x;
  const int j = blockIdx.x;
  const int c = blockIdx.y * 64 + tid;
  const int b = blockIdx.z;
  const float w = decay[c] * (1.0f / Tt);
  const float u = first[c] * (1.0f / Tt);
  const size_t bbase = (size_t)b * Tt * Cc;

  // forward carry for chunk j (dir 0)
  size_t fb = (((size_t)b * NCH + j) * 3) * Cc + c;
  float pf = sum[fb], qf = sum[fb + Cc], of_ = sum[fb + 2 * Cc];
  for (int s = 0; s < CHUNK; ++s) {
    lds[(s * 3 + 0) * 64 + tid] = pf;
    lds[(s * 3 + 1) * 64 + tid] = qf;
    lds[(s * 3 + 2) * 64 + tid] = of_;
    int t = seq2mem(j * CHUNK + s, perm);
    size_t idx = bbase + (size_t)t * Cc + c;
    float kt = k[idx], vt = v[idx];
    float no = fmaxf(w + of_, kt);
    float A = __expf(w + of_ - no);
    float Bc = __expf(kt - no);
    pf = A * pf + Bc * vt;
    qf = A * qf + Bc;
    of_ = no;
  }

  // backward carry: backward-chunk index is NCH-1-j (dir 1)
  size_t bbk = ((((size_t)1 * Bb + b) * NCH + (NCH - 1 - j)) * 3) * Cc + c;
  float pb = sum[bbk], qb = sum[bbk + Cc], ob = sum[bbk + 2 * Cc];
  for (int s = CHUNK - 1; s >= 0; --s) {
    int t = seq2mem(j * CHUNK + s, perm);
    size_t idx = bbase + (size_t)t * Cc + c;
    float kt = k[idx], vt = v[idx];
    float pF = lds[(s * 3 + 0) * 64 + tid];
    float qF = lds[(s * 3 + 1) * 64 + tid];
    float oF = lds[(s * 3 + 2) * 64 + tid];
    float cur = u + kt;
    float m = fmaxf(fmaxf(oF, ob), cur);
    float ef = __expf(oF - m), eb = __expf(ob - m), ec = __expf(cur - m);
    float y = (ef * pF + eb * pb + ec * vt) / (ef * qF + eb * qb + ec);
    v[idx] = y; // in-place: this position is read only by this thread
    float no = fmaxf(w + ob, kt);
    float A = __expf(w + ob - no);
    float Bc = __expf(kt - no);
    pb = A * pb + Bc * vt;
    qb = A * qb + Bc;
    ob = no;
  }
}

// ---------------------------------------------------------------------------
// Launcher
// ---------------------------------------------------------------------------
extern "C" void kernel_launch(void* const* d_in, const int* in_sizes, int n_in,
                              void* d_out, int out_size, void* d_ws, size_t ws_size,
                              hipStream_t stream) {
  (void)in_sizes; (void)n_in; (void)out_size; (void)ws_size;
  const float* x     = (const float*)d_in[0];
  const float* w1    = (const float*)d_in[1];
  const float* w3    = (const float*)d_in[2];
  const float* w5    = (const float*)d_in[3];
  const float* alpha = (const float*)d_in[4];
  const float* Wk    = (const float*)d_in[5];
  const float* Wv    = (const float*)d_in[6];
  const float* Wr    = (const float*)d_in[7];
  const float* Wo    = (const float*)d_in[8];
  const float* sd    = (const float*)d_in[9];   // spatial_decay [2,C]
  const float* sf    = (const float*)d_in[10];  // spatial_first [2,C]
  float* out = (float*)d_out;
  float* ws  = (float*)d_ws;

  const size_t N1 = (size_t)Bb * Tt * Cc; // 6,291,456 floats
  float* xs   = ws;
  float* kb   = ws + N1;
  float* vb   = ws + 2 * N1;
  float* srb  = ws + 3 * N1;
  float* weff = ws + 4 * N1;        // 25*C floats
  float* sum  = weff + 25 * Cc;     // 2*B*NCH*3*C floats (~2.4 MB)

  // 1) OmniShift (folded 5x5 depthwise conv)
  build_weff<<<dim3((25 * Cc + 191) / 192), dim3(192), 0, stream>>>(w1, w3, w5, alpha, weff);
  omnishift<<<dim3(Bb * Tt), dim3(Cc), 0, stream>>>(x, weff, xs);

  // 2) k/v/r projections (fp32 WMMA, fused sigmoid)
  gemm_qkv<<<dim3((Bb * Tt) / 16, Cc / 16), dim3(32), 0, stream>>>(xs, Wk, Wv, Wr, kb, vb, srb);

  // 3) Two bidirectional WKV scans (row-major, then column-major via perm)
  for (int j = 0; j < 2; ++j) {
    const float* decay = sd + j * Cc;
    const float* first = sf + j * Cc;
    wkv_pass1<<<dim3(NCH, Bb, 2), dim3(Cc), 0, stream>>>(kb, vb, decay, sum, j);
    wkv_pass2<<<dim3(Bb, 2), dim3(Cc), 0, stream>>>(decay, sum);
    wkv_pass3<<<dim3(NCH, Cc / 64, Bb), dim3(64), 0, stream>>>(kb, vb, decay, first, sum, j);
  }

  // 4) Output projection: (sr .* v) @ Wo^T  (fp32 WMMA)
  gemm_out<<<dim3((Bb * Tt) / 16, Cc / 16), dim3(32), 0, stream>>>(srb, vb, Wo, out);
}